// ColumnParallelLinearWithMoE_28922309771636
// MI455X (gfx1250) — compile-verified
//
#include <hip/hip_runtime.h>

#define E_    8
#define BS_   64
#define S_    1024
#define D_    1024
#define OPP_  512
#define OUTC_ 4096

typedef float v2f __attribute__((ext_vector_type(2)));
typedef float v8f __attribute__((ext_vector_type(8)));

struct alignas(16) F4 { float x, y, z, w; };

#define KT     16
#define APITCH 20   // 16 K-floats + 4 pad: rows 16B-aligned, LDS-bank conflict-free

#if __has_builtin(__builtin_amdgcn_global_load_async_to_lds_b128) && \
    __has_builtin(__builtin_amdgcn_s_wait_asynccnt)
#define USE_ASYNC_LDS 1
// param types per hipcc diagnostic: int __attribute__((vector_size(16))) in AS1 / AS3
typedef int v4i_vs __attribute__((vector_size(16)));
typedef __attribute__((address_space(1))) v4i_vs glob_v4i;
typedef __attribute__((address_space(3))) v4i_vs shar_v4i;
#else
#define USE_ASYNC_LDS 0
#endif

// --- build expert-of-row table: inv[row] = expert ---
__global__ void moe_inv_kernel(const long long* __restrict__ idx, int* __restrict__ inv) {
  int t = threadIdx.x;
  if (t < E_ * (BS_ / E_)) inv[(int)idx[t]] = t >> 3;  // idx_list[e][p], flat t = e*8+p
}

// --- zero the 7/8 of out not owned by the row's expert (float4 stores) ---
__global__ void moe_zero_kernel(float* __restrict__ out, const int* __restrict__ inv) {
  size_t i4 = (size_t)blockIdx.x * blockDim.x + threadIdx.x; // float4 index
  int row = (int)(i4 >> 20);          // 2^20 float4 per batch row (1024*4096/4)
  int cb  = (int)(i4 >> 7) & 7;       // 512-channel block id
  if (cb != inv[row]) {
    F4 z = {0.f, 0.f, 0.f, 0.f};
    *(F4*)(out + i4 * 4) = z;
  }
}

// --- grouped GEMM via V_WMMA_F32_16X16X4_F32, async-LDS double buffering ---
__global__ __launch_bounds__(256) void moe_gemm_wmma(
    const float* __restrict__ x, const long long* __restrict__ idx,
    const float* __restrict__ W, const float* __restrict__ bvec,
    float* __restrict__ out)
{
  __shared__ float As[2][128 * APITCH];
  __shared__ float Bs[2][128 * APITCH];

  const int tid   = threadIdx.x;
  const int lane  = tid & 31;
  const int lm    = lane & 15;
  const int khalf = (lane >> 4) << 1;  // 0 or 2: A/B fragment K half per ISA layout
  const int hi8   = (lane >> 4) << 3;  // C/D: lanes 16-31 hold M+8
  const int w     = tid >> 5;
  const int wyM   = (w & 1) * 64;      // wave M offset in 128-tile
  const int wxN   = (w >> 1) * 32;     // wave N offset in 128-tile

  const int e  = blockIdx.z;           // expert
  const int bN = blockIdx.x;           // 0..3   (N tile of 128)
  const int bM = blockIdx.y;           // 0..63  (M tile of 128)
  const int p     = bM >> 3;           // which of the 8 routed rows
  const int sBase = (bM & 7) * 128;    // seq offset (M-tile divides S)
  const int r     = (int)idx[e * 8 + p];

  const float* aBase = x + ((size_t)r * S_ + sBase) * D_;
  const float* bBase = W + ((size_t)e * OPP_ + (size_t)bN * 128) * D_;

  const int ldRow = tid >> 2;          // 0..63 (two passes cover 128 rows)
  const int ldK4  = (tid & 3) * 4;     // float4 slot within K-tile

  v8f acc[4][2];
  const v8f vzero = {0.f,0.f,0.f,0.f,0.f,0.f,0.f,0.f};
#pragma unroll
  for (int i = 0; i < 4; ++i)
#pragma unroll
    for (int j = 0; j < 2; ++j) acc[i][j] = vzero;

  const int NKT = D_ / KT;  // 64

#if USE_ASYNC_LDS
  // ---------- ASYNCcnt-pipelined global -> LDS staging (no VGPR data) ----------
  // each thread issues 4 async B128 loads per K-tile (2 for A, 2 for B)
  {
    // prologue: tile 0 -> buffer 0
#pragma unroll
    for (int i = 0; i < 2; ++i) {
      const int row = ldRow + i * 64;
      __builtin_amdgcn_global_load_async_to_lds_b128(
          (glob_v4i*)(aBase + (size_t)row * D_ + ldK4),
          (shar_v4i*)&As[0][row * APITCH + ldK4], 0, 0);
      __builtin_amdgcn_global_load_async_to_lds_b128(
          (glob_v4i*)(bBase + (size_t)row * D_ + ldK4),
          (shar_v4i*)&Bs[0][row * APITCH + ldK4], 0, 0);
    }
  }
  for (int kt = 0; kt < NKT; ++kt) {
    const int buf = kt & 1;
    if (kt + 1 < NKT) {  // issue next tile into other buffer
      const int k0 = (kt + 1) * KT;
      const int nb = buf ^ 1;
#pragma unroll
      for (int i = 0; i < 2; ++i) {
        const int row = ldRow + i * 64;
        __builtin_amdgcn_global_load_async_to_lds_b128(
            (glob_v4i*)(aBase + (size_t)row * D_ + k0 + ldK4),
            (shar_v4i*)&As[nb][row * APITCH + ldK4], 0, 0);
        __builtin_amdgcn_global_load_async_to_lds_b128(
            (glob_v4i*)(bBase + (size_t)row * D_ + k0 + ldK4),
            (shar_v4i*)&Bs[nb][row * APITCH + ldK4], 0, 0);
      }
      __builtin_amdgcn_s_wait_asynccnt(4);  // current tile landed; next still in flight
    } else {
      __builtin_amdgcn_s_wait_asynccnt(0);
    }
    __syncthreads();  // all waves' async writes to buf visible

#pragma unroll
    for (int kk = 0; kk < KT; kk += 4) {
      v2f af[4], bf[2];
#pragma unroll
      for (int i = 0; i < 4; ++i)
        af[i] = *(const v2f*)&As[buf][(wyM + i * 16 + lm) * APITCH + kk + khalf];
#pragma unroll
      for (int j = 0; j < 2; ++j)
        bf[j] = *(const v2f*)&Bs[buf][(wxN + j * 16 + lm) * APITCH + kk + khalf];
#pragma unroll
      for (int i = 0; i < 4; ++i)
#pragma unroll
        for (int j = 0; j < 2; ++j)
          acc[i][j] = __builtin_amdgcn_wmma_f32_16x16x4_f32(
              false, af[i], false, bf[j], (short)0, acc[i][j], false, false);
    }
    __syncthreads();  // buf fully consumed before it is overwritten next iter
  }
#else
  // ---------- fallback: register-staged double buffering ----------
  F4 aR[2], bR[2];
#pragma unroll
  for (int i = 0; i < 2; ++i) {
    int row = ldRow + i * 64;
    aR[i] = *(const F4*)(aBase + (size_t)row * D_ + ldK4);
    bR[i] = *(const F4*)(bBase + (size_t)row * D_ + ldK4);
  }
#pragma unroll
  for (int i = 0; i < 2; ++i) {
    int row = ldRow + i * 64;
    *(F4*)&As[0][row * APITCH + ldK4] = aR[i];
    *(F4*)&Bs[0][row * APITCH + ldK4] = bR[i];
  }
  __syncthreads();

  for (int kt = 0; kt < NKT; ++kt) {
    const int buf = kt & 1;
    if (kt + 1 < NKT) {
      const int k0 = (kt + 1) * KT;
#pragma unroll
      for (int i = 0; i < 2; ++i) {
        int row = ldRow + i * 64;
        aR[i] = *(const F4*)(aBase + (size_t)row * D_ + k0 + ldK4);
        bR[i] = *(const F4*)(bBase + (size_t)row * D_ + k0 + ldK4);
      }
    }
#pragma unroll
    for (int kk = 0; kk < KT; kk += 4) {
      v2f af[4], bf[2];
#pragma unroll
      for (int i = 0; i < 4; ++i)
        af[i] = *(const v2f*)&As[buf][(wyM + i * 16 + lm) * APITCH + kk + khalf];
#pragma unroll
      for (int j = 0; j < 2; ++j)
        bf[j] = *(const v2f*)&Bs[buf][(wxN + j * 16 + lm) * APITCH + kk + khalf];
#pragma unroll
      for (int i = 0; i < 4; ++i)
#pragma unroll
        for (int j = 0; j < 2; ++j)
          acc[i][j] = __builtin_amdgcn_wmma_f32_16x16x4_f32(
              false, af[i], false, bf[j], (short)0, acc[i][j], false, false);
    }
    if (kt + 1 < NKT) {
      const int nb = buf ^ 1;
#pragma unroll
      for (int i = 0; i < 2; ++i) {
        int row = ldRow + i * 64;
        *(F4*)&As[nb][row * APITCH + ldK4] = aR[i];
        *(F4*)&Bs[nb][row * APITCH + ldK4] = bR[i];
      }
    }
    __syncthreads();
  }
#endif

  // epilogue: bias add + scatter into out[r, s, e*512 + o]
  const int oColBase = e * OPP_ + bN * 128 + wxN;  // bias index == out channel index
#pragma unroll
  for (int i = 0; i < 4; ++i) {
#pragma unroll
    for (int j = 0; j < 2; ++j) {
      const int n = oColBase + j * 16 + lm;
      const float bias = bvec[n];
#pragma unroll
      for (int v = 0; v < 8; ++v) {
        const int mloc = wyM + i * 16 + v + hi8;
        const size_t off = ((size_t)r * S_ + sBase + mloc) * (size_t)OUTC_ + n;
        out[off] = acc[i][j][v] + bias;
      }
    }
  }
}

extern "C" void kernel_launch(void* const* d_in, const int* in_sizes, int n_in,
                              void* d_out, int out_size, void* d_ws, size_t ws_size,
                              hipStream_t stream) {
  const float*     x   = (const float*)d_in[0];
  const long long* idx = (const long long*)d_in[1];
  const float*     Wp  = (const float*)d_in[2];
  const float*     bp  = (const float*)d_in[3];
  float*           out = (float*)d_out;
  int*             inv = (int*)d_ws;

  (void)in_sizes; (void)n_in; (void)out_size; (void)ws_size;

  moe_inv_kernel<<<1, 64, 0, stream>>>(idx, inv);

  const size_t total_f4 = (size_t)BS_ * S_ * OUTC_ / 4;  // 67,108,864
  moe_zero_kernel<<<(unsigned)(total_f4 / 256), 256, 0, stream>>>(out, inv);

  dim3 grid(OPP_ / 128, (S_ * (BS_ / E_)) / 128, E_);    // (4, 64, 8)
  moe_gemm_wmma<<<grid, 256, 0, stream>>>(x, idx, Wp, bp, out);
}